// Panorama_34660386078785
// MI455X (gfx1250) — compile-verified
//
#include <hip/hip_runtime.h>

// ---------------------------------------------------------------------------
// Panorama warp + composite, MI455X (gfx1250).
// Canvas tiles are streamed into LDS by the Tensor Data Mover (TDM) while the
// VMEM pipe handles the irregular bilinear gathers from the frame.
// ---------------------------------------------------------------------------

typedef __attribute__((ext_vector_type(4))) unsigned int uint32x4_t_;
typedef __attribute__((ext_vector_type(8))) int          int32x8_t_;
typedef __attribute__((ext_vector_type(4))) int          int32x4_t_;
typedef __attribute__((ext_vector_type(4))) float        floatx4;

#define HC 2048
#define WC 3072
#define HF 1080
#define WF 1920
#define FPLANE (HF * WF)      // frame channel-plane stride
#define CPLANE (HC * WC)      // canvas/out channel-plane stride
#define TW 64                 // tile width  (pixels)
#define THT 16                // tile height (pixels)

__global__ __launch_bounds__(256)
void pano_warp_composite(const float* __restrict__ frame,
                         const float* __restrict__ Hmat,
                         const float* __restrict__ canvas,
                         float* __restrict__ out)
{
    // 4 channels x 16 rows x 64 cols of canvas, filled by TDM (x,y order)
    __shared__ __align__(16) float tile[4 * THT * TW];
    __shared__ float sH[9];

    const int tid = threadIdx.x;
    const int gx0 = blockIdx.x * TW;   // tile origin in canvas
    const int gy0 = blockIdx.y * THT;

    if (tid < 32) {                    // wave 0 only (wave32)
        if (tid == 0) {
            // 3x3 inverse of Hmat (adjugate / det), once per block
            float a0 = Hmat[0], a1 = Hmat[1], a2 = Hmat[2];
            float a3 = Hmat[3], a4 = Hmat[4], a5 = Hmat[5];
            float a6 = Hmat[6], a7 = Hmat[7], a8 = Hmat[8];
            float c00 = a4 * a8 - a5 * a7;
            float c01 = a2 * a7 - a1 * a8;
            float c02 = a1 * a5 - a2 * a4;
            float c10 = a5 * a6 - a3 * a8;
            float c11 = a0 * a8 - a2 * a6;
            float c12 = a2 * a3 - a0 * a5;
            float c20 = a3 * a7 - a4 * a6;
            float c21 = a1 * a6 - a0 * a7;
            float c22 = a0 * a4 - a1 * a3;
            float r = 1.0f / (a0 * c00 + a1 * c10 + a2 * c20);
            sH[0] = c00 * r; sH[1] = c01 * r; sH[2] = c02 * r;
            sH[3] = c10 * r; sH[4] = c11 * r; sH[5] = c12 * r;
            sH[6] = c20 * r; sH[7] = c21 * r; sH[8] = c22 * r;
        }

        // ---- Tensor DMA descriptors (D#), per CDNA5 ISA --
        // One 2D tile (TW x THT) per channel plane; for 2D tensors D# groups
        // 2/3 are unused, so all trailing builtin args are zero-filled.
        const unsigned long long s0 = (unsigned long long)WC; // row stride (elems)
        int32x4_t_ gz4;  gz4[0] = 0; gz4[1] = 0; gz4[2] = 0; gz4[3] = 0;
        int32x8_t_ gz8;
        gz8[0] = 0; gz8[1] = 0; gz8[2] = 0; gz8[3] = 0;
        gz8[4] = 0; gz8[5] = 0; gz8[6] = 0; gz8[7] = 0;

#pragma unroll
        for (int c = 0; c < 4; ++c) {
            unsigned long long gaddr = (unsigned long long)(uintptr_t)
                (canvas + (size_t)c * CPLANE + (size_t)gy0 * WC + gx0);
            unsigned int lds_off = (unsigned int)(uintptr_t)&tile[c * THT * TW];

            // Group 0: count=1 | lds_addr | global_addr[56:0] | type=2 (bits 127:126)
            uint32x4_t_ g0;
            g0[0] = 1u;                                    // count=1, user descriptor
            g0[1] = lds_off;                               // LDS byte address
            g0[2] = (unsigned int)(gaddr & 0xFFFFFFFFull); // global_addr[31:0] -> bits 95:64
            g0[3] = (unsigned int)((gaddr >> 32) & 0x01FFFFFFull) | 0x80000000u; // addr[56:32] | type=2

            // Group 1 (256-bit, little-endian words):
            // data_size=2 (4B); tensor_dim0=WC @bit48; tensor_dim1=HC @bit80;
            // tile_dim0=TW @bit112; tile_dim1=THT @bit128; tile_dim2=0 (unused);
            // tensor_dim0_stride=WC @bit160; tensor_dim1_stride unused (2D).
            int32x8_t_ g1;
            g1[0] = (int)(2u << 16);                                   // data_size = 4 bytes
            g1[1] = (int)(((unsigned)WC & 0xFFFFu) << 16);             // dim0 lo16
            g1[2] = (int)((((unsigned)WC >> 16) & 0xFFFFu) |
                          (((unsigned)HC & 0xFFFFu) << 16));           // dim0 hi16 | dim1 lo16
            g1[3] = (int)((((unsigned)HC >> 16) & 0xFFFFu) |
                          ((unsigned)TW << 16));                       // dim1 hi16 | tile_dim0
            g1[4] = (int)((unsigned)THT);                              // tile_dim1 | tile_dim2=0
            g1[5] = (int)(unsigned)(s0 & 0xFFFFFFFFull);               // stride0[31:0]
            g1[6] = (int)(unsigned)((s0 >> 32) & 0xFFFFull);           // stride0[47:32] | stride1 lo16=0
            g1[7] = 0;                                                 // stride1 hi bits = 0

            __builtin_amdgcn_tensor_load_to_lds(g0, g1, gz4, gz4, gz8, 0);
        }
        __builtin_amdgcn_s_wait_tensorcnt(0);
    }
    __syncthreads();   // publish tile + sH to all 8 waves

    const float h0 = sH[0], h1 = sH[1], h2 = sH[2];
    const float h3 = sH[3], h4 = sH[4], h5 = sH[5];
    const float h6 = sH[6], h7 = sH[7], h8 = sH[8];

    const int tx4 = (tid & (TW / 4 - 1)) * 4;  // 0..60, float4 column
    const int ty  = tid >> 4;                  // 0..15
    const float yf = (float)(gy0 + ty);
    const float xf = (float)(gx0 + tx4);

    // Incremental homography across the 4 pixels of this thread
    float nx = fmaf(h0, xf, fmaf(h1, yf, h2));
    float ny = fmaf(h3, xf, fmaf(h4, yf, h5));
    float nw = fmaf(h6, xf, fmaf(h7, yf, h8));

    float wr[4][4];   // [pixel][channel] warped samples
#pragma unroll
    for (int px = 0; px < 4; ++px) {
        float invw = 1.0f / nw;
        float sx = nx * invw;
        float sy = ny * invw;
        float x0f = floorf(sx), y0f = floorf(sy);
        float fx = sx - x0f,   fy = sy - y0f;
        int x0 = (int)x0f, y0 = (int)y0f;
        int x1 = x0 + 1,   y1 = y0 + 1;
        // in-bounds masks (reference zeroes OOB taps)
        float mx0 = (x0 >= 0 && x0 < WF) ? 1.0f : 0.0f;
        float mx1 = (x1 >= 0 && x1 < WF) ? 1.0f : 0.0f;
        float my0 = (y0 >= 0 && y0 < HF) ? 1.0f : 0.0f;
        float my1 = (y1 >= 0 && y1 < HF) ? 1.0f : 0.0f;
        int xc0 = min(max(x0, 0), WF - 1), xc1 = min(max(x1, 0), WF - 1);
        int yc0 = min(max(y0, 0), HF - 1), yc1 = min(max(y1, 0), HF - 1);
        float w00 = (1.0f - fx) * (1.0f - fy) * mx0 * my0;
        float w01 = fx * (1.0f - fy) * mx1 * my0;
        float w10 = (1.0f - fx) * fy * mx0 * my1;
        float w11 = fx * fy * mx1 * my1;
        int b00 = yc0 * WF + xc0, b01 = yc0 * WF + xc1;
        int b10 = yc1 * WF + xc0, b11 = yc1 * WF + xc1;
#pragma unroll
        for (int c = 0; c < 4; ++c) {
            const float* fp = frame + c * FPLANE;
            wr[px][c] = fmaf(w00, fp[b00],
                        fmaf(w01, fp[b01],
                        fmaf(w10, fp[b10], w11 * fp[b11])));
        }
        nx += h0; ny += h3; nw += h6;
    }

    // Canvas tile from LDS (ds_load_b128 per channel)
    floatx4 cv[4];
#pragma unroll
    for (int c = 0; c < 4; ++c)
        cv[c] = *(const floatx4*)&tile[(c * THT + ty) * TW + tx4];

    // Alpha-over composite
    floatx4 o[4];
#pragma unroll
    for (int px = 0; px < 4; ++px) {
        float a_s = wr[px][3];
        float a_c = cv[3][px];
        float k = a_c * (1.0f - a_s);
        o[0][px] = fmaf(wr[px][0], a_s, cv[0][px] * k);
        o[1][px] = fmaf(wr[px][1], a_s, cv[1][px] * k);
        o[2][px] = fmaf(wr[px][2], a_s, cv[2][px] * k);
        o[3][px] = a_s + k;
    }

    // Streaming (non-temporal) B128 stores — keep frame resident in L2
    size_t obase = (size_t)(gy0 + ty) * WC + (size_t)(gx0 + tx4);
#pragma unroll
    for (int c = 0; c < 4; ++c)
        __builtin_nontemporal_store(o[c], (floatx4*)(out + (size_t)c * CPLANE + obase));
}

extern "C" void kernel_launch(void* const* d_in, const int* in_sizes, int n_in,
                              void* d_out, int out_size, void* d_ws, size_t ws_size,
                              hipStream_t stream) {
    const float* frame  = (const float*)d_in[0];  // (1,4,1080,1920)
    const float* Hmat   = (const float*)d_in[1];  // (3,3)
    const float* canvas = (const float*)d_in[2];  // (1,4,2048,3072)
    float* out = (float*)d_out;                   // (1,4,2048,3072)

    dim3 grid(WC / TW, HC / THT);                 // 48 x 128 tiles
    pano_warp_composite<<<grid, 256, 0, stream>>>(frame, Hmat, canvas, out);
}